// IDWT1D_33148557590648
// MI455X (gfx1250) — compile-verified
//
#include <hip/hip_runtime.h>

// Inverse 1D Haar DWT (2-tap reconstruction), fully streaming:
//   out2[i] = ( lo1*a + hi1*d , lo0*a + hi0*d )  where in2[i] = (a, d)
// Memory-bound (0.37 flop/byte): floor ~23us at 23.3 TB/s.
// Data path: gfx1250 async bulk loads (global_load_async_to_lds_b128 th:NT,
// ASYNCcnt double-buffered per wave) -> ds_load_b128 -> VALU FMA ->
// non-temporal global_store_b128. Both streams are read/written exactly once
// (268 MB each > 192 MB L2), so both carry NT temporal hints.

typedef float v4f __attribute__((ext_vector_type(4)));

__global__ __launch_bounds__(256) void idwt_haar_async_kernel(
    const v4f* __restrict__ in, const float* __restrict__ lo,
    const float* __restrict__ hi, v4f* __restrict__ out, unsigned n4)
{
    // 8 waves/block * 2 buffers * 32 lanes * 16B = 8 KB
    __shared__ v4f ldsbuf[512];

    // Uniform filter taps (scalar loads).
    const float l0 = lo[0], l1 = lo[1];
    const float h0 = hi[0], h1 = hi[1];

    const unsigned tid   = threadIdx.x;
    const unsigned lane  = tid & 31u;
    const unsigned wib   = tid >> 5;                          // wave in block
    const unsigned gwave = ((blockIdx.x * blockDim.x) + tid) >> 5;
    const unsigned nwave = (gridDim.x * blockDim.x) >> 5;
    const unsigned ntile = n4 >> 5;                           // tiles of 32 float4

    // Per-lane LDS byte offsets of the two buffer slots (low 32 bits of the
    // flat address of a __shared__ object == its LDS offset).
    const unsigned slot0    = wib * 64u + lane;
    const unsigned lds_off0 = (unsigned)(size_t)&ldsbuf[slot0];
    const unsigned lds_off1 = (unsigned)(size_t)&ldsbuf[slot0 + 32u];

    unsigned t = gwave;
    if (t < ntile) {
        // Prime buffer 0 with this wave's first tile.
        {
            const v4f* g = in + ((size_t)t * 32u + lane);
            asm volatile("global_load_async_to_lds_b128 %0, %1, off th:TH_LOAD_NT"
                         :: "v"(lds_off0), "v"(g) : "memory");
        }
        unsigned buf = 0u;
        while (t < ntile) {
            const unsigned tn = t + nwave;
            if (tn < ntile) {
                // WAR guard: previous ds_load from the buffer we are about to
                // overwrite must have completed (async LDS writes are
                // unordered vs DS reads).
                asm volatile("s_wait_dscnt 0x0" ::: "memory");
                const v4f*     gn   = in + ((size_t)tn * 32u + lane);
                const unsigned offn = buf ? lds_off0 : lds_off1;
                asm volatile("global_load_async_to_lds_b128 %0, %1, off th:TH_LOAD_NT"
                             :: "v"(offn), "v"(gn) : "memory");
                // Two loads in flight; async loads complete in order, so
                // asynccnt<=1 means the older (current) tile has landed.
                asm volatile("s_wait_asynccnt 0x1" ::: "memory");
            } else {
                // Last tile: only it is outstanding.
                asm volatile("s_wait_asynccnt 0x0" ::: "memory");
            }

            const v4f xv = ldsbuf[slot0 + (buf ? 32u : 0u)];  // ds_load_b128
            v4f r;
            r.x = l1 * xv.x + h1 * xv.y;   // s*(a+d)
            r.y = l0 * xv.x + h0 * xv.y;   // s*(a-d)
            r.z = l1 * xv.z + h1 * xv.w;
            r.w = l0 * xv.z + h0 * xv.w;
            // Write-once output: keep it out of L2 (streaming, 268MB >> 192MB).
            __builtin_nontemporal_store(r, out + ((size_t)t * 32u + lane));

            buf ^= 1u;
            t = tn;
        }
    }

    // Tail (n4 not a multiple of 32) — never taken for the actual shapes,
    // kept for robustness. Direct load/compute/store, NT both ways.
    const unsigned gthread = blockIdx.x * blockDim.x + tid;
    const unsigned nthread = gridDim.x * blockDim.x;
    for (unsigned i = (ntile << 5) + gthread; i < n4; i += nthread) {
        const v4f xv = __builtin_nontemporal_load(in + i);
        v4f r;
        r.x = l1 * xv.x + h1 * xv.y;
        r.y = l0 * xv.x + h0 * xv.y;
        r.z = l1 * xv.z + h1 * xv.w;
        r.w = l0 * xv.z + h0 * xv.w;
        __builtin_nontemporal_store(r, out + i);
    }
}

extern "C" void kernel_launch(void* const* d_in, const int* in_sizes, int n_in,
                              void* d_out, int out_size, void* d_ws, size_t ws_size,
                              hipStream_t stream) {
    const float* x  = (const float*)d_in[0];   // (64, 524288, 2) f32
    const float* lo = (const float*)d_in[1];   // (2,) f32
    const float* hi = (const float*)d_in[2];   // (2,) f32
    float* out = (float*)d_out;                // (64, 1048576, 1) f32

    const unsigned n4 = (unsigned)(in_sizes[0] / 4);  // 16,777,216 float4s

    dim3 block(256);
    dim3 grid(2048);   // 16384 waves; 32 tiles/wave, double-buffered async
    idwt_haar_async_kernel<<<grid, block, 0, stream>>>(
        (const v4f*)x, lo, hi, (v4f*)out, n4);
}